// AttentionContext_20486994002266
// MI455X (gfx1250) — compile-verified
//
#include <hip/hip_runtime.h>

typedef float v2f __attribute__((ext_vector_type(2)));
typedef float v8f __attribute__((ext_vector_type(8)));

// ---------------------------------------------------------------------------
// Generic fp32 WMMA GEMM:  C[m,n] = alpha * sum_k A[m,k]*B[k,n] + bias[n]
// A row-major (lda). B row-major KxN (BT=false) or stored NxK (BT=true, ldb =
// per-row k-stride). One wave per 16x16 C tile, K-loop in steps of 4 using
// V_WMMA_F32_16X16X4_F32.  grid = (Ntiles, Mtiles, heads).
//
// Fragment layouts per CDNA5 ISA 7.12.2:
//   A 16x4 : lanes 0-15 rows M=0..15, vgpr0/1 = K=0/1 ; lanes 16-31 = K=2/3
//   B 4x16 : vgpr0 = rows K=0 (lanes0-15) / K=2 (lanes16-31), vgpr1 = K=1/K=3
//   C 16x16: vgpr r = row (r + 8*half), col = lane&15
// ---------------------------------------------------------------------------
template <bool BT>
__global__ __launch_bounds__(32)
void wmma_gemm_f32(const float* __restrict__ A, int lda, long hsA,
                   const float* __restrict__ B, int ldb, long hsB,
                   const float* __restrict__ bias, long hsBias,
                   float* __restrict__ C, int ldc, long hsC,
                   int K, float alpha)
{
    const int h = blockIdx.z;
    A += (long)h * hsA;
    B += (long)h * hsB;
    C += (long)h * hsC;
    if (bias) bias += (long)h * hsBias;

    const int lane = threadIdx.x;
    const int half = lane >> 4;      // 0: K pair {0,1}, 1: K pair {2,3}
    const int lr   = lane & 15;
    const int mrow = blockIdx.y * 16 + lr;
    const int ncol = blockIdx.x * 16 + lr;

    v8f acc = {};
    for (int d = 0; d < K; d += 4) {
        const int kb = d + half * 2;
        v2f a, b;
        a.x = A[(long)mrow * lda + kb];
        a.y = A[(long)mrow * lda + kb + 1];
        if (BT) {
            b.x = B[(long)ncol * ldb + kb];
            b.y = B[(long)ncol * ldb + kb + 1];
        } else {
            b.x = B[(long)kb * ldb + ncol];
            b.y = B[(long)(kb + 1) * ldb + ncol];
        }
        acc = __builtin_amdgcn_wmma_f32_16x16x4_f32(
            /*neg_a=*/false, a, /*neg_b=*/false, b,
            /*c_mod=*/(short)0, acc, /*reuse_a=*/false, /*reuse_b=*/false);
    }

    const float bv = bias ? bias[ncol] : 0.0f;
#pragma unroll
    for (int r = 0; r < 8; ++r) {
        const int row = blockIdx.y * 16 + r + half * 8;
        C[(long)row * ldc + ncol] = alpha * acc[r] + bv;
    }
}

// ---------------------------------------------------------------------------
// Fused attention, partial pass.  grid = (B=64, USPLIT=4), 256 threads.
// Block (b, split) streams enc[b, split*512 .. +512, :] exactly once.
// Each of the 8 waves owns 64 consecutive u values for ALL 4 heads, keeping
// flash-style online-softmax state (m, s) and the 4x512 ctx accumulator in
// VGPRs (lane owns e = 128*j + 4*lane + c).  Waves merge through LDS; block
// writes un-normalized partial (M, S, ctx) for the final combine kernel.
// enc bias term (phi_s . psi_b) is constant over u -> softmax invariant ->
// omitted; the 1/sqrt(K) scale is pre-folded into q.
// ---------------------------------------------------------------------------
__global__ __launch_bounds__(256)
void attn_partial(const float* __restrict__ enc, const float* __restrict__ q,
                  float* __restrict__ pctx, float* __restrict__ pm,
                  float* __restrict__ ps)
{
    const int b     = blockIdx.x;
    const int split = blockIdx.y;
    const int lane  = threadIdx.x & 31;
    const int wave  = threadIdx.x >> 5;

    __shared__ float s_ctx[4 * 512];
    __shared__ float s_m[8][4];
    __shared__ float s_s[8][4];
    __shared__ float s_M[4];

    for (int i = threadIdx.x; i < 2048; i += 256) s_ctx[i] = 0.0f;

    float4 qr[4][4], acc[4][4];
    float  m[4], s[4];
#pragma unroll
    for (int h = 0; h < 4; ++h) {
        m[h] = -__builtin_inff();
        s[h] = 0.0f;
#pragma unroll
        for (int j = 0; j < 4; ++j) {
            qr[h][j]  = *(const float4*)(q + ((long)(b * 4 + h)) * 512 + 128 * j + 4 * lane);
            acc[h][j] = make_float4(0.0f, 0.0f, 0.0f, 0.0f);
        }
    }

    const int u0 = split * 512 + wave * 64;
    for (int t = 0; t < 64; ++t) {
        const float* row = enc + ((long)b * 2048 + u0 + t) * 512;
        float4 ev[4];
#pragma unroll
        for (int j = 0; j < 4; ++j) ev[j] = *(const float4*)(row + 128 * j + 4 * lane);

        float e[4];
#pragma unroll
        for (int h = 0; h < 4; ++h) {
            float p = 0.0f;
#pragma unroll
            for (int j = 0; j < 4; ++j) {
                p = fmaf(qr[h][j].x, ev[j].x, p);
                p = fmaf(qr[h][j].y, ev[j].y, p);
                p = fmaf(qr[h][j].z, ev[j].z, p);
                p = fmaf(qr[h][j].w, ev[j].w, p);
            }
#pragma unroll
            for (int off = 16; off >= 1; off >>= 1) p += __shfl_xor(p, off, 32);
            e[h] = p;   // uniform across the wave after butterfly
        }

#pragma unroll
        for (int h = 0; h < 4; ++h) {
            if (e[h] <= m[h]) {            // common path: no max update, no rescale
                const float w = __expf(e[h] - m[h]);
                s[h] += w;
#pragma unroll
                for (int j = 0; j < 4; ++j) {
                    acc[h][j].x = fmaf(w, ev[j].x, acc[h][j].x);
                    acc[h][j].y = fmaf(w, ev[j].y, acc[h][j].y);
                    acc[h][j].z = fmaf(w, ev[j].z, acc[h][j].z);
                    acc[h][j].w = fmaf(w, ev[j].w, acc[h][j].w);
                }
            } else {                        // new running max: rescale state
                const float corr = __expf(m[h] - e[h]);   // exp(-inf)=0 on 1st hit
                s[h] = fmaf(s[h], corr, 1.0f);
#pragma unroll
                for (int j = 0; j < 4; ++j) {
                    acc[h][j].x = fmaf(acc[h][j].x, corr, ev[j].x);
                    acc[h][j].y = fmaf(acc[h][j].y, corr, ev[j].y);
                    acc[h][j].z = fmaf(acc[h][j].z, corr, ev[j].z);
                    acc[h][j].w = fmaf(acc[h][j].w, corr, ev[j].w);
                }
                m[h] = e[h];
            }
        }
    }

    if (lane == 0) {
#pragma unroll
        for (int h = 0; h < 4; ++h) { s_m[wave][h] = m[h]; s_s[wave][h] = s[h]; }
    }
    __syncthreads();

    if (threadIdx.x < 4) {               // per-head block max + rebased sum
        const int h = threadIdx.x;
        float M = -__builtin_inff();
#pragma unroll
        for (int w = 0; w < 8; ++w) M = fmaxf(M, s_m[w][h]);
        float S = 0.0f;
#pragma unroll
        for (int w = 0; w < 8; ++w) S += __expf(s_m[w][h] - M) * s_s[w][h];
        s_M[h] = M;
        pm[((long)b * 4 + split) * 4 + h] = M;
        ps[((long)b * 4 + split) * 4 + h] = S;
    }
    __syncthreads();

#pragma unroll
    for (int h = 0; h < 4; ++h) {
        const float f = __expf(m[h] - s_M[h]);   // wave-uniform
#pragma unroll
        for (int j = 0; j < 4; ++j) {
            const int base = h * 512 + 128 * j + 4 * lane;
            atomicAdd(&s_ctx[base + 0], f * acc[h][j].x);
            atomicAdd(&s_ctx[base + 1], f * acc[h][j].y);
            atomicAdd(&s_ctx[base + 2], f * acc[h][j].z);
            atomicAdd(&s_ctx[base + 3], f * acc[h][j].w);
        }
    }
    __syncthreads();

    float* dst = pctx + ((long)(b * 4 + split)) * 2048;
    for (int i = threadIdx.x; i < 2048; i += 256) dst[i] = s_ctx[i];
}

// ---------------------------------------------------------------------------
// Combine the 4 U-splits (flash merge), divide by softmax denom, l2-normalize
// over E=512, write combined[b, h*512 + e].  grid = (64, 4), 128 threads.
// ---------------------------------------------------------------------------
__global__ __launch_bounds__(128)
void attn_combine(const float* __restrict__ pctx, const float* __restrict__ pm,
                  const float* __restrict__ ps, float* __restrict__ combined)
{
    const int b = blockIdx.x;
    const int h = blockIdx.y;
    const int t = threadIdx.x;
    __shared__ float red[128];

    float M = -__builtin_inff();
#pragma unroll
    for (int sp = 0; sp < 4; ++sp) M = fmaxf(M, pm[((long)b * 4 + sp) * 4 + h]);
    float S = 0.0f, f[4];
#pragma unroll
    for (int sp = 0; sp < 4; ++sp) {
        f[sp] = __expf(pm[((long)b * 4 + sp) * 4 + h] - M);
        S += f[sp] * ps[((long)b * 4 + sp) * 4 + h];
    }
    const float invS = 1.0f / S;

    float c[4];
    float sq = 0.0f;
#pragma unroll
    for (int i = 0; i < 4; ++i) {
        const int e = t + 128 * i;
        float v = 0.0f;
#pragma unroll
        for (int sp = 0; sp < 4; ++sp)
            v = fmaf(f[sp], pctx[((long)(b * 4 + sp)) * 2048 + h * 512 + e], v);
        c[i] = v * invS;
        sq = fmaf(c[i], c[i], sq);
    }
    red[t] = sq;
    __syncthreads();
    for (int st = 64; st > 0; st >>= 1) {
        if (t < st) red[t] += red[t + st];
        __syncthreads();
    }
    const float rn = rsqrtf(fmaxf(red[0], 1e-12f));
#pragma unroll
    for (int i = 0; i < 4; ++i)
        combined[(long)b * 2048 + h * 512 + t + 128 * i] = c[i] * rn;
}

// ---------------------------------------------------------------------------
extern "C" void kernel_launch(void* const* d_in, const int* in_sizes, int n_in,
                              void* d_out, int out_size, void* d_ws, size_t ws_size,
                              hipStream_t stream)
{
    (void)in_sizes; (void)n_in; (void)out_size; (void)ws_size;

    const float* dec   = (const float*)d_in[0];  // (64, 1024)
    const float* enc   = (const float*)d_in[1];  // (64, 2048, 512)
    const float* phi_w = (const float*)d_in[2];  // (4, 1024, 256)
    const float* phi_b = (const float*)d_in[3];  // (4, 256)
    const float* psi_w = (const float*)d_in[4];  // (4, 512, 256)
    const float* psi_b = (const float*)d_in[5];  // (4, 256) -- softmax-invariant, unused
    const float* out_w = (const float*)d_in[6];  // (2048, 1024)
    const float* out_b = (const float*)d_in[7];  // (1024,)
    (void)psi_b;
    float* out = (float*)d_out;                  // (64, 1024)

    float* ws       = (float*)d_ws;              // ~3.3 MB of fp32 scratch
    float* phi_s    = ws;                        // 65536   (B,H,K)
    float* qbuf     = ws + 65536;                // 131072  (B,H,E)
    float* combined = ws + 196608;               // 131072  (B, H*E)
    float* pctx     = ws + 327680;               // 524288  (B,4,H,E)
    float* pm       = ws + 851968;               // 1024    (B,4,H)
    float* ps       = ws + 852992;               // 1024    (B,4,H)

    // 1) phi_s[b,h,k] = dec @ phi_w[h] + phi_b[h]          (M=64,N=256,K=1024 x4 heads)
    wmma_gemm_f32<false><<<dim3(16, 4, 4), 32, 0, stream>>>(
        dec, 1024, 0L, phi_w, 256, 262144L, phi_b, 256L,
        phi_s, 1024, 256L, 1024, 1.0f);

    // 2) q[b,h,e] = (1/sqrt(256)) * phi_s[b,h] @ psi_w[h]^T (M=64,N=512,K=256 x4 heads)
    wmma_gemm_f32<true><<<dim3(32, 4, 4), 32, 0, stream>>>(
        phi_s, 1024, 256L, psi_w, 256, 131072L, nullptr, 0L,
        qbuf, 2048, 512L, 256, 0.0625f);

    // 3) streaming fused attention over enc (single read of 256 MB)
    attn_partial<<<dim3(64, 4), 256, 0, stream>>>(enc, qbuf, pctx, pm, ps);
    attn_combine<<<dim3(64, 4), 128, 0, stream>>>(pctx, pm, ps, combined);

    // 4) out = combined @ out_w + out_b                    (M=64,N=1024,K=2048)
    wmma_gemm_f32<false><<<dim3(64, 4, 1), 32, 0, stream>>>(
        combined, 2048, 0L, out_w, 1024, 0L, out_b, 0L,
        out, 1024, 0L, 2048, 1.0f);
}